// GCN_62912680952090
// MI455X (gfx1250) — compile-verified
//
#include <hip/hip_runtime.h>
#include <hip/hip_bf16.h>

typedef __attribute__((ext_vector_type(2))) float v2f;
typedef __attribute__((ext_vector_type(8))) float v8f;

// ---------------------------------------------------------------------------
// Generic tiled WMMA f32 GEMM:  D = epilogue( Aop @ Bop )
//   Block tile: 64 (M) x 128 (N), K-step 32. 256 threads = 8 waves (wave32).
//   Wave grid 2x4, each wave computes a 32x32 tile = 2x2 WMMA 16x16 subtiles.
//
// AMODE: 0: a = A[m*ldA + k]
//        1: a = A[m*ldA + k] * scaleK[k]          (T * diag(v);  A2 * 1/colmax)
//        2: a = A[k*ldA + m] * scaleK[k]          (T^T * diag(v), gc2 mult)
//        3: a = max(A[m*ldA + k], 0)              (relu(Z) on the fly)
// BMODE: 0: b = (n < Nb) ? B[k*ldB + n] : 0       (plain row-major K x N)
//        1: b = B[n*ldB + k]                       (B = T^T, read T directly)
// EPI:   0: store acc
//        1: store relu(acc + bias[n])
//        2: store acc + bias[n]
//        3: store ((m==n)?1:acc) * mask[m*ldMask+n]   (eye-blend + adjacency)
// ---------------------------------------------------------------------------
template <int AMODE, int BMODE, int EPI>
__global__ __launch_bounds__(256) void gemm_wmma_f32(
    const float* __restrict__ A, const float* __restrict__ B,
    float* __restrict__ D, const float* __restrict__ scaleK,
    const float* __restrict__ bias, const float* __restrict__ mask,
    int M, int N, int K, int ldA, int ldB, int ldD, int Nb, int Nd, int ldMask) {
  constexpr int BM = 64, BN = 128, BK = 32, LDSP = BK + 2;  // pad: stride 34
  __shared__ float As[BM][LDSP];   // A tile, [m][k]
  __shared__ float Bs[BN][LDSP];   // B tile transposed, [n][k]

  const int tid = threadIdx.x;
  const int lane = tid & 31;
  const int li = lane & 15;        // row/col index within 16
  const int lh = lane >> 4;        // 0/1 -> k-pair selector
  const int w = tid >> 5;          // wave id 0..7
  const int m_off = (w >> 2) * 32; // wave tile origin inside block tile
  const int n_off = (w & 3) * 32;
  const int m0 = blockIdx.y * BM;
  const int n0 = blockIdx.x * BN;

  v8f acc[2][2] = {};

  for (int kt = 0; kt < K; kt += BK) {
    // ---- stage A tile into LDS -------------------------------------------
    for (int idx = tid; idx < BM * BK; idx += 256) {
      int r, c;
      float a;
      if (AMODE == 2) {
        c = idx / BM; r = idx % BM;  // coalesced along m
        a = A[(size_t)(kt + c) * ldA + (m0 + r)] * scaleK[kt + c];
      } else {
        r = idx / BK; c = idx % BK;  // coalesced along k
        a = A[(size_t)(m0 + r) * ldA + (kt + c)];
        if (AMODE == 1) a *= scaleK[kt + c];
        if (AMODE == 3) a = fmaxf(a, 0.0f);
      }
      As[r][c] = a;
    }
    // ---- stage B tile (transposed) into LDS ------------------------------
    for (int idx = tid; idx < BN * BK; idx += 256) {
      int r, c;
      float b;
      if (BMODE == 1) {
        r = idx / BK; c = idx % BK;  // coalesced along k
        b = B[(size_t)(n0 + r) * ldB + (kt + c)];
      } else {
        c = idx / BN; r = idx % BN;  // coalesced along n
        int ng = n0 + r;
        b = (ng < Nb) ? B[(size_t)(kt + c) * ldB + ng] : 0.0f;
      }
      Bs[r][c] = b;
    }
    __syncthreads();

    // ---- WMMA inner loop --------------------------------------------------
#pragma unroll
    for (int kk = 0; kk < BK; kk += 4) {
      v2f af[2], bf[2];
#pragma unroll
      for (int i = 0; i < 2; ++i) {
        const float* pa = &As[m_off + i * 16 + li][kk + lh * 2];
        af[i] = v2f{pa[0], pa[1]};
      }
#pragma unroll
      for (int j = 0; j < 2; ++j) {
        const float* pb = &Bs[n_off + j * 16 + li][kk + lh * 2];
        bf[j] = v2f{pb[0], pb[1]};
      }
#pragma unroll
      for (int i = 0; i < 2; ++i)
#pragma unroll
        for (int j = 0; j < 2; ++j)
          acc[i][j] = __builtin_amdgcn_wmma_f32_16x16x4_f32(
              false, af[i], false, bf[j], (short)0, acc[i][j], false, false);
    }
    __syncthreads();
  }

  // ---- epilogue + store ---------------------------------------------------
#pragma unroll
  for (int i = 0; i < 2; ++i) {
#pragma unroll
    for (int j = 0; j < 2; ++j) {
      const int ng = n0 + n_off + j * 16 + li;
      if (ng < Nd) {
#pragma unroll
        for (int r = 0; r < 8; ++r) {
          const int mg = m0 + m_off + i * 16 + lh * 8 + r;
          float v = acc[i][j][r];
          if (EPI == 1) v = fmaxf(v + bias[ng], 0.0f);
          else if (EPI == 2) v = v + bias[ng];
          else if (EPI == 3) v = ((mg == ng) ? 1.0f : v) * mask[(size_t)mg * ldMask + ng];
          D[(size_t)mg * ldD + ng] = v;
        }
      }
    }
  }
}

// v[i] = sum_c H[i*ldH + c] * p[c]
__global__ void matvec_kernel(const float* __restrict__ H, const float* __restrict__ p,
                              float* __restrict__ v, int rows, int C, int ldH) {
  int i = blockIdx.x * blockDim.x + threadIdx.x;
  if (i < rows) {
    float s = 0.0f;
    for (int c = 0; c < C; ++c) s += H[(size_t)i * ldH + c] * p[c];
    v[i] = s;
  }
}

// inv[f] = 1 / max_e A[e*E + f]   (max over the full dense column, zeros incl.)
__global__ void colmax_inv_kernel(const float* __restrict__ A, float* __restrict__ inv, int E) {
  int f = blockIdx.x * blockDim.x + threadIdx.x;
  if (f < E) {
    float m = A[f];
    for (int e = 1; e < E; ++e) m = fmaxf(m, A[(size_t)e * E + f]);
    inv[f] = 1.0f / m;
  }
}

extern "C" void kernel_launch(void* const* d_in, const int* in_sizes, int n_in,
                              void* d_out, int out_size, void* d_ws, size_t ws_size,
                              hipStream_t stream) {
  constexpr int N = 1024, E = 4096, FV = 512, FE = 64, NH = 256, FOUT = 250;
  constexpr int F2 = 2 * FE;  // 128
  (void)in_sizes; (void)n_in; (void)out_size; (void)ws_size;

  const float* X     = (const float*)d_in[0];   // (N, FV)
  const float* Z     = (const float*)d_in[1];   // (E, FE)
  const float* adj_e = (const float*)d_in[2];   // (E, E)
  const float* adj_v = (const float*)d_in[3];   // (N, N)
  const float* T     = (const float*)d_in[4];   // (N, E)
  const float* w1    = (const float*)d_in[5];   // (FV, NH)
  const float* p1    = (const float*)d_in[6];   // (FE,)
  const float* b1    = (const float*)d_in[7];   // (NH,)
  const float* w2    = (const float*)d_in[8];   // (FE, F2)
  const float* p2    = (const float*)d_in[9];   // (NH,)
  const float* b2    = (const float*)d_in[10];  // (F2,)
  const float* w3    = (const float*)d_in[11];  // (NH, FOUT)
  const float* p3    = (const float*)d_in[12];  // (F2,)
  const float* b3    = (const float*)d_in[13];  // (FOUT,)

  float* ws = (float*)d_ws;
  float* v1   = ws;                 // 4096 (reused for v3)
  float* XW1  = v1   + E;           // N*NH = 262144 (reused for xW3, padded 256)
  float* A1   = XW1  + N * NH;      // N*N  = 1048576 (reused for A3)
  float* xbuf = A1   + N * N;       // N*NH = 262144
  float* ZW2  = xbuf + N * NH;      // E*F2 = 524288
  float* v2   = ZW2  + E * F2;      // 1024 (padded slot 4096)
  float* inv  = v2   + E;           // 4096
  float* A2   = inv  + E;           // E*E = 16777216  (~75.5 MB total f32)

  float* x_out = (float*)d_out;             // (N, FOUT) row-major
  float* z_out = x_out + N * FOUT;          // (E, F2)

  dim3 blk(256);

  // ===== gc1 (node layer) ================================================
  // v1 = Z @ p1
  matvec_kernel<<<dim3((E + 255) / 256), blk, 0, stream>>>(Z, p1, v1, E, FE, FE);
  // XW1 = X @ w1                      M=1024 N=256 K=512
  gemm_wmma_f32<0, 0, 0><<<dim3(NH / 128, N / 64), blk, 0, stream>>>(
      X, w1, XW1, nullptr, nullptr, nullptr, N, NH, FV, FV, NH, NH, NH, NH, 0);
  // A1 = (eye + (1-eye)*(T diag(v1) T^T)) * adj_v     M=N=1024 K=4096
  gemm_wmma_f32<1, 1, 3><<<dim3(N / 128, N / 64), blk, 0, stream>>>(
      T, T, A1, v1, nullptr, adj_v, N, N, E, E, E, N, N, N, N);
  // x = relu(A1 @ XW1 + b1)           M=1024 N=256 K=1024
  gemm_wmma_f32<0, 0, 1><<<dim3(NH / 128, N / 64), blk, 0, stream>>>(
      A1, XW1, xbuf, nullptr, b1, nullptr, N, NH, N, N, NH, NH, NH, NH, 0);

  // ===== gc2 (edge layer) ================================================
  // ZW2 = relu(Z) @ w2                M=4096 N=128 K=64
  gemm_wmma_f32<3, 0, 0><<<dim3(F2 / 128, E / 64), blk, 0, stream>>>(
      Z, w2, ZW2, nullptr, nullptr, nullptr, E, F2, FE, FE, F2, F2, F2, F2, 0);
  // v2 = x @ p2
  matvec_kernel<<<dim3((N + 255) / 256), blk, 0, stream>>>(xbuf, p2, v2, N, NH, NH);
  // A2 = (eye + (1-eye)*(T^T diag(v2) T)) * adj_e     M=N=4096 K=1024
  gemm_wmma_f32<2, 0, 3><<<dim3(E / 128, E / 64), blk, 0, stream>>>(
      T, T, A2, v2, nullptr, adj_e, E, E, N, E, E, E, E, E, E);
  // inv[f] = 1 / colmax_e A2[e,f]
  colmax_inv_kernel<<<dim3((E + 255) / 256), blk, 0, stream>>>(A2, inv, E);
  // z = relu((A2 * inv[k]) @ ZW2 + b2)   M=4096 N=128 K=4096  -> d_out tail
  gemm_wmma_f32<1, 0, 1><<<dim3(F2 / 128, E / 64), blk, 0, stream>>>(
      A2, ZW2, z_out, inv, b2, nullptr, E, F2, E, E, F2, F2, F2, F2, 0);

  // ===== gc3 (node layer, no relu) =======================================
  // v3 = z @ p3   (reuse v1 slot)
  matvec_kernel<<<dim3((E + 255) / 256), blk, 0, stream>>>(z_out, p3, v1, E, F2, F2);
  // A3 = (eye + (1-eye)*(T diag(v3) T^T)) * adj_v   (reuse A1 slot)
  gemm_wmma_f32<1, 1, 3><<<dim3(N / 128, N / 64), blk, 0, stream>>>(
      T, T, A1, v1, nullptr, adj_v, N, N, E, E, E, N, N, N, N);
  // xW3 = x @ w3  (pad FOUT 250 -> 256; B-loads guarded, pad cols become 0)
  gemm_wmma_f32<0, 0, 0><<<dim3(256 / 128, N / 64), blk, 0, stream>>>(
      xbuf, w3, XW1, nullptr, nullptr, nullptr, N, 256, NH, NH, FOUT, 256, FOUT, 256, 0);
  // x_out = A3 @ xW3 + b3             M=1024 N=250(store-guarded) K=1024
  gemm_wmma_f32<0, 0, 2><<<dim3(256 / 128, N / 64), blk, 0, stream>>>(
      A1, XW1, x_out, nullptr, b3, nullptr, N, 256, N, N, 256, FOUT, 256, FOUT, 0);
}